// ConvolutionalCapsule3D_68238440399051
// MI455X (gfx1250) — compile-verified
//
#include <hip/hip_runtime.h>
#include <math.h>

typedef __attribute__((ext_vector_type(2))) float v2f;
typedef __attribute__((ext_vector_type(8))) float v8f;

#define EPSR 1e-7f
#define NB 4
#define ND 4
#define NH 14
#define NW 14
#define NCH 32
#define NI (ND * NH * NW)    /* 784   */
#define CI (NI * NCH)        /* 25088 */
#define NJ 24
#define NM 16
#define ROWT ((NI * 4) / 16) /* 196 row tiles of the (n_i*4, 4) A matrix */

__device__ __forceinline__ float coord_add(int io, int q) {
  // coordinate-addition value for m = 12+q (q=1,2,3), 0 for q==0
  if (q == 0) return 0.f;
  const int iw = io % NW;
  const int ih = (io / NW) % NH;
  const int id = io / (NW * NH);
  if (q == 1) return (float)(id - ND / 2);
  if (q == 2) return (float)(ih - NH / 2);
  return (float)(iw - NW / 2);
}

// ---------------------------------------------------------------------------
// Kernel 1: votes via V_WMMA_F32_16X16X4_F32.
// Per (b,c): GEMM  A(n_i*4, 4) x B(4, 96) -> votes(n_i*4, 96), K=4.
// One wave per (b, c, row-tile); each wave runs 6 WMMAs (6 column tiles).
// All per-lane invariants (q, the two i positions, coord-add values, output
// base pointers) hoisted out of the ct loop; stores use immediate offsets.
// ---------------------------------------------------------------------------
__global__ __launch_bounds__(256)
void votes_wmma_kernel(const float* __restrict__ pose,
                       const float* __restrict__ wmat,
                       float* __restrict__ votes) {
  const int lane = threadIdx.x & 31;
  const int wave = (blockIdx.x * blockDim.x + threadIdx.x) >> 5;

  const int rt = wave % ROWT;
  const int bc = wave / ROWT;
  const int c  = bc % NCH;
  const int b  = bc / NCH;
  if (b >= NB) return;

  const int hi  = lane >> 4;  // 0 = lanes 0-15, 1 = lanes 16-31
  const int l16 = lane & 15;
  const int k0  = hi * 2;     // K values held by this half-wave

  // A fragment: row R = rt*16 + l16, this lane holds (k0, k0+1).
  const int R  = rt * 16 + l16;
  const int ia = R >> 2;
  const int p  = R & 3;
  const float* ua =
      pose + ((((size_t)b * NI + ia) * NCH + c) << 4) + p * 4 + k0;
  v2f afrag;
  afrag.x = ua[0];
  afrag.y = ua[1];

  // Per-lane invariants for B fragment / D store.
  const int q  = l16 & 3;   // output pose column, invariant across ct
  const int jl = l16 >> 2;  // j offset within a column tile
  const int io0 = rt * 4 + hi * 2;  // i position for D regs 0..3
  const int io1 = io0 + 1;          // i position for D regs 4..7
  const float cadd0 = coord_add(io0, q);
  const float cadd1 = coord_add(io1, q);
  float* out0 =
      votes + (((size_t)b * CI + (size_t)io0 * NCH + c) * NJ + jl) * NM + q;
  float* out1 =
      votes + (((size_t)b * CI + (size_t)io1 * NCH + c) * NJ + jl) * NM + q;
  const float* wb = wmat + (((size_t)c * NJ + jl) << 4) + q;

#pragma unroll
  for (int ct = 0; ct < 6; ++ct) {
    // B fragment: column n = l16 of tile ct -> j = ct*4 + jl, rows k0,k0+1.
    v2f bfrag;
    bfrag.x = wb[ct * 64 + k0 * 4];
    bfrag.y = wb[ct * 64 + (k0 + 1) * 4];

    v8f acc = {};
    // (neg_a, A, neg_b, B, c_mod, C, reuse_a, reuse_b)
    acc = __builtin_amdgcn_wmma_f32_16x16x4_f32(false, afrag, false, bfrag,
                                                (short)0, acc, false, false);

    // D: reg v -> (io0, po=v) for v<4, (io1, po=v-4) for v>=4; m = po*4+q.
#pragma unroll
    for (int v = 0; v < 4; ++v) {
      float val = acc[v];
      if (v == 3) val += cadd0;  // po==3 => coordinate addition (0 if q==0)
      out0[ct * 64 + v * 4] = val;
    }
#pragma unroll
    for (int v = 4; v < 8; ++v) {
      float val = acc[v];
      if (v == 7) val += cadd1;
      out1[ct * 64 + (v - 4) * 4] = val;
    }
  }
}

// ---------------------------------------------------------------------------
__global__ void zero_kernel(float* __restrict__ p, int n) {
  int i = blockIdx.x * blockDim.x + threadIdx.x;
  if (i < n) p[i] = 0.0f;
}

// ---------------------------------------------------------------------------
// Accumulation pass (init or E-step). One wave32 per input capsule ci; lane j
// (<24) owns output capsule j. mean / 1/(2s+EPS) hoisted to registers; S0/S1/
// S2 accumulated in lane-private registers across the ci loop, flushed once
// per wave to LDS (ds_add_f32), then once per block to global atomics.
// ---------------------------------------------------------------------------
__global__ __launch_bounds__(256)
void ems_acc_kernel(const float* __restrict__ votes,
                    const float* __restrict__ act,
                    const float* __restrict__ meanS,
                    const float* __restrict__ svarS,
                    const float* __restrict__ coefS,
                    const float* __restrict__ logaS,
                    float* __restrict__ S0,
                    float* __restrict__ S1,
                    float* __restrict__ S2,
                    int estep) {
  __shared__ float lS0[NJ];
  __shared__ float lS1[NJ * NM];
  __shared__ float lS2[NJ * NM];
  for (int t = threadIdx.x; t < NJ; t += blockDim.x) lS0[t] = 0.f;
  for (int t = threadIdx.x; t < NJ * NM; t += blockDim.x) {
    lS1[t] = 0.f;
    lS2[t] = 0.f;
  }
  __syncthreads();

  const int b    = blockIdx.y;
  const int j    = threadIdx.x & 31;
  const int wpg  = gridDim.x * (blockDim.x >> 5);
  const int wave = blockIdx.x * (blockDim.x >> 5) + (threadIdx.x >> 5);

  // Loop-invariant per-lane Gaussian parameters (E-step only).
  float mu[NM], i2s[NM];
  float lcc = 0.f;
  if (estep && j < NJ) {
#pragma unroll
    for (int m = 0; m < NM; ++m) {
      mu[m]  = meanS[(b * NJ + j) * NM + m];
      i2s[m] = 1.f / (2.f * svarS[(b * NJ + j) * NM + m] + EPSR);
    }
    lcc = logaS[b * NJ + j] + coefS[b * NJ + j];
  } else {
#pragma unroll
    for (int m = 0; m < NM; ++m) { mu[m] = 0.f; i2s[m] = 0.f; }
  }

  // Lane-private moment accumulators.
  float aS0 = 0.f, aS1[NM], aS2[NM];
#pragma unroll
  for (int m = 0; m < NM; ++m) { aS1[m] = 0.f; aS2[m] = 0.f; }

  for (int ci = wave; ci < CI; ci += wpg) {
    float ai = act[(size_t)b * CI + ci];
    ai = (ai >= 0.f) ? ai : 0.f;  // ROUTE_MIN clamp

    float vv[NM];
    if (j < NJ) {
      const float4* vp = (const float4*)(votes +
          ((size_t)b * CI + ci) * (NJ * NM) + j * NM);
      const float4 t0 = vp[0], t1 = vp[1], t2 = vp[2], t3 = vp[3];
      vv[0] = t0.x;  vv[1] = t0.y;  vv[2] = t0.z;  vv[3] = t0.w;
      vv[4] = t1.x;  vv[5] = t1.y;  vv[6] = t1.z;  vv[7] = t1.w;
      vv[8] = t2.x;  vv[9] = t2.y;  vv[10] = t2.z; vv[11] = t2.w;
      vv[12] = t3.x; vv[13] = t3.y; vv[14] = t3.z; vv[15] = t3.w;
    } else {
#pragma unroll
      for (int m = 0; m < NM; ++m) vv[m] = 0.f;
    }

    float r;
    if (!estep) {
      r = ai * (1.0f / NJ);
    } else {
      float logap = -1e30f;
      if (j < NJ) {
        float expo = 0.f;
#pragma unroll
        for (int m = 0; m < NM; ++m) {
          const float d = vv[m] - mu[m];
          expo += d * d * i2s[m];
        }
        logap = lcc - expo;
      }
      float mx = logap;
#pragma unroll
      for (int off = 16; off >= 1; off >>= 1)
        mx = fmaxf(mx, __shfl_xor(mx, off, 32));
      float e = (j < NJ) ? __expf(logap - mx) : 0.f;
      float ssum = e;
#pragma unroll
      for (int off = 16; off >= 1; off >>= 1) ssum += __shfl_xor(ssum, off, 32);
      r = (e / ssum) * ai;
    }

    aS0 += r;
#pragma unroll
    for (int m = 0; m < NM; ++m) {
      const float rv = r * vv[m];
      aS1[m] += rv;
      aS2[m] += rv * vv[m];
    }
  }

  // Flush lane-private accumulators: one LDS atomic batch per wave.
  if (j < NJ) {
    atomicAdd(&lS0[j], aS0);
#pragma unroll
    for (int m = 0; m < NM; ++m) {
      atomicAdd(&lS1[j * NM + m], aS1[m]);
      atomicAdd(&lS2[j * NM + m], aS2[m]);
    }
  }
  __syncthreads();
  for (int t = threadIdx.x; t < NJ; t += blockDim.x)
    atomicAdd(&S0[b * NJ + t], lS0[t]);
  for (int t = threadIdx.x; t < NJ * NM; t += blockDim.x) {
    atomicAdd(&S1[b * NJ * NM + t], lS1[t]);
    atomicAdd(&S2[b * NJ * NM + t], lS2[t]);
  }
}

// ---------------------------------------------------------------------------
// M-step: one block per batch, 384 threads = (j, m). Intra-16-lane shuffle
// reductions over m; cost stats over j via a tiny LDS array.
// ---------------------------------------------------------------------------
__global__ __launch_bounds__(NJ* NM)
void ems_mstep_kernel(const float* __restrict__ S0,
                      const float* __restrict__ S1,
                      const float* __restrict__ S2,
                      const float* __restrict__ beta_v,
                      const float* __restrict__ beta_a,
                      float* __restrict__ meanS,
                      float* __restrict__ svarS,
                      float* __restrict__ coefS,
                      float* __restrict__ logaS,
                      float inv_temp,
                      float* __restrict__ out_mean,
                      float* __restrict__ out_act) {
  const int b = blockIdx.x;
  const int t = threadIdx.x;
  const int j = t >> 4;
  const int m = t & 15;
  __shared__ float cst[NJ];

  const float s0  = S0[b * NJ + j];
  const float den = s0 + EPSR;
  const float s1  = S1[(b * NJ + j) * NM + m];
  const float s2  = S2[(b * NJ + j) * NM + m];
  const float mu  = s1 / (den + EPSR);
  float sv = (s2 - 2.f * mu * s1 + mu * mu * s0) / (den + EPSR);
  sv = fmaxf(sv, 0.f);

  meanS[(b * NJ + j) * NM + m] = mu;
  svarS[(b * NJ + j) * NM + m] = sv;
  out_mean[(b * NJ + j) * NM + m] = mu;

  float costm = (beta_v[j * NM + m] + logf(sqrtf(sv + EPSR) + EPSR)) * den;
  float coefm = logf(2.f * 3.14159265358979f * sv + EPSR);
#pragma unroll
  for (int off = 8; off >= 1; off >>= 1) {  // stays within 16-lane m-group
    costm += __shfl_xor(costm, off, 32);
    coefm += __shfl_xor(coefm, off, 32);
  }
  if (m == 0) cst[j] = costm;
  __syncthreads();

  float cm = 0.f;
  for (int jj = 0; jj < NJ; ++jj) cm += cst[jj];
  cm *= (1.f / NJ);
  float var = 0.f;
  for (int jj = 0; jj < NJ; ++jj) {
    const float d = cst[jj] - cm;
    var += d * d;
  }
  const float cs = sqrtf(var * (1.f / NJ) + EPSR);

  const float z  = inv_temp * (beta_a[j] + (cm - costm) / (cs + EPSR));
  const float aj = 1.f / (1.f + __expf(-z));
  if (m == 0) {
    coefS[b * NJ + j] = -0.5f * coefm;
    logaS[b * NJ + j] = logf(aj + EPSR);
    out_act[b * NJ + j] = aj;
  }
}

// ---------------------------------------------------------------------------
extern "C" void kernel_launch(void* const* d_in, const int* in_sizes, int n_in,
                              void* d_out, int out_size, void* d_ws,
                              size_t ws_size, hipStream_t stream) {
  (void)in_sizes; (void)n_in; (void)out_size; (void)ws_size;

  const float* pose   = (const float*)d_in[0];
  const float* act    = (const float*)d_in[1];
  const float* wmat   = (const float*)d_in[2];
  const float* beta_v = (const float*)d_in[3];
  const float* beta_a = (const float*)d_in[4];

  float* ws = (float*)d_ws;
  const size_t VOTES = (size_t)NB * CI * NJ * NM;  // ~154 MB (fits in L2)
  float* votes = ws;
  float* S0    = votes + VOTES;      // NB*NJ
  float* S1    = S0 + NB * NJ;       // NB*NJ*NM
  float* S2    = S1 + NB * NJ * NM;  // NB*NJ*NM
  float* meanS = S2 + NB * NJ * NM;
  float* svarS = meanS + NB * NJ * NM;
  float* coefS = svarS + NB * NJ * NM;
  float* logaS = coefS + NB * NJ;

  float* out_mean = (float*)d_out;
  float* out_act  = out_mean + NB * NJ * NM;

  // 1) votes einsum via WMMA (K=4 fp32).
  {
    const int totalWaves = NB * NCH * ROWT;  // 25088 waves
    const int threads = 256;
    const int blocks = (totalWaves * 32 + threads - 1) / threads;  // 3136
    votes_wmma_kernel<<<blocks, threads, 0, stream>>>(pose, wmat, votes);
  }

  const int ACCN = NB * (NJ + 2 * NJ * NM);  // S0,S1,S2 contiguous = 3168
  dim3 accGrid(32, NB);

  // 2) init stats (uniform r scaled by a_i), first activations at temp 0.5.
  zero_kernel<<<(ACCN + 255) / 256, 256, 0, stream>>>(S0, ACCN);
  ems_acc_kernel<<<accGrid, 256, 0, stream>>>(votes, act, meanS, svarS, coefS,
                                              logaS, S0, S1, S2, 0);
  ems_mstep_kernel<<<NB, NJ * NM, 0, stream>>>(S0, S1, S2, beta_v, beta_a,
                                               meanS, svarS, coefS, logaS,
                                               0.5f, out_mean, out_act);

  // 3) two routing iterations (counters 1,2 -> temps 0.6, 0.8).
  float it = 0.5f;
  for (int cnt = 1; cnt <= 2; ++cnt) {
    it += 0.1f * (float)cnt;
    zero_kernel<<<(ACCN + 255) / 256, 256, 0, stream>>>(S0, ACCN);
    ems_acc_kernel<<<accGrid, 256, 0, stream>>>(votes, act, meanS, svarS,
                                                coefS, logaS, S0, S1, S2, 1);
    ems_mstep_kernel<<<NB, NJ * NM, 0, stream>>>(S0, S1, S2, beta_v, beta_a,
                                                 meanS, svarS, coefS, logaS,
                                                 it, out_mean, out_act);
  }
}